// NonlinearElectronicEmbedding_43550968382197
// MI455X (gfx1250) — compile-verified
//
#include <hip/hip_runtime.h>

typedef float v2f __attribute__((ext_vector_type(2)));
typedef float v4f __attribute__((ext_vector_type(4)));
typedef float v8f __attribute__((ext_vector_type(8)));

#define FDIM 256

__device__ __forceinline__ float fast_rcp(float x) {
  return __builtin_amdgcn_rcpf(x);   // v_rcp_f32, ~1 ulp
}
__device__ __forceinline__ float silu1702(float x) {
  // x * sigmoid(1.702 x) == x / (1 + exp(-1.702 x)); fast rcp avoids IEEE div chain
  return x * fast_rcp(1.0f + __expf(-1.702f * x));
}
// order-preserving float <-> uint mapping for atomic max (0u == minimum)
__device__ __forceinline__ unsigned f2ord(float f) {
  unsigned u = __float_as_uint(f);
  return (u & 0x80000000u) ? ~u : (u | 0x80000000u);
}
__device__ __forceinline__ float ord2f(unsigned u) {
  return __uint_as_float((u & 0x80000000u) ? (u ^ 0x80000000u) : ~u);
}

__global__ void init_seg_kernel(unsigned* __restrict__ segmax, float* __restrict__ segsum, int B) {
  int i = blockIdx.x * blockDim.x + threadIdx.x;
  if (i < B) { segmax[i] = 0u; segsum[i] = 0.0f; }
}

// Kin[b,f] = E[b]*Wkf[f] + bkf[f] ; Vin[b,f] = E[b]*Wvf[f]
__global__ void mol_embed_kernel(const float* __restrict__ E,
                                 const float* __restrict__ Wkf,
                                 const float* __restrict__ bkf,
                                 const float* __restrict__ Wvf,
                                 float* __restrict__ Kin,
                                 float* __restrict__ Vin, int total) {
  int i = blockIdx.x * blockDim.x + threadIdx.x;
  if (i >= total) return;
  int b = i >> 8, f = i & 255;
  float e = E[b];
  Kin[i] = e * Wkf[f] + bkf[f];
  Vin[i] = e * Wvf[f];
}

// C = [act_out]( [act_in](A) @ op(W) + bias + res ), all 256-wide, M rows.
// TRANSB=false: C[i,n] = sum_k A'[i,k] * W[n,k]   (i.e. @ W.T, row-major W)
// TRANSB=true : C[i,n] = sum_k A'[i,k] * W[k,n]   (i.e. @ W)
// One 16x16 C tile per wave via V_WMMA_F32_16X16X4_F32, K-loop of 64.
template <bool ACT_IN, bool ACT_OUT, bool TRANSB, bool HAS_BIAS, bool HAS_RES>
__global__ void gemm256_wmma_kernel(const float* __restrict__ A,
                                    const float* __restrict__ W,
                                    const float* __restrict__ bias,
                                    const float* __restrict__ res,
                                    float* __restrict__ C, int M) {
  const int lane = threadIdx.x & 31;
  const int wid  = threadIdx.x >> 5;
  const int tile = blockIdx.x * 8 + wid;
  const int tiles_n = FDIM / 16;
  const int tm = tile / tiles_n, tn = tile % tiles_n;
  if (tm * 16 >= M) return;
  const int lh = lane >> 4;   // which K-pair half this lane holds
  const int lm = lane & 15;
  const int arow = tm * 16 + lm;   // A fragment: lane -> row M
  const int bcol = tn * 16 + lm;   // B fragment: lane -> col N
  v8f acc = {};
  for (int k = 0; k < FDIM; k += 4) {
    const int ka = k + 2 * lh;
    // A 16x4 f32 layout: VGPR0/1 = K={0,1} (lanes 0-15) / K={2,3} (lanes 16-31)
    v2f a = *(const v2f*)(A + arow * FDIM + ka);
    if (ACT_IN) { a.x = silu1702(a.x); a.y = silu1702(a.y); }
    v2f b;
    if (!TRANSB) {
      b = *(const v2f*)(W + bcol * FDIM + ka);          // B[k][n] = W[n][k]
    } else {
      b.x = W[ka * FDIM + bcol];                        // B[k][n] = W[k][n]
      b.y = W[(ka + 1) * FDIM + bcol];
    }
    acc = __builtin_amdgcn_wmma_f32_16x16x4_f32(false, a, false, b,
                                                (short)0, acc, false, false);
  }
  // C layout: VGPR r -> M = r (lanes 0-15) / M = r+8 (lanes 16-31); lane -> N
  #pragma unroll
  for (int r = 0; r < 8; ++r) {
    const int row = tm * 16 + r + 8 * lh;
    float c = acc[r];
    if (HAS_BIAS) c += bias[bcol];
    if (HAS_RES)  c += res[row * FDIM + bcol];
    if (ACT_OUT)  c = silu1702(c);
    C[row * FDIM + bcol] = c;
  }
}

// One wave32 per row: dot[n] = x[n] . kp[seg[n]] ; segment atomic max.
__global__ void dot_max_kernel(const float* __restrict__ x,
                               const float* __restrict__ kp,
                               const int* __restrict__ seg,
                               float* __restrict__ dot,
                               unsigned* __restrict__ segmax, int N) {
  const int row = blockIdx.x * 8 + (threadIdx.x >> 5);
  if (row >= N) return;
  const int lane = threadIdx.x & 31;
  const int s = seg[row];
  const v4f* xr = (const v4f*)(x + (size_t)row * FDIM);
  const v4f* kr = (const v4f*)(kp + (size_t)s * FDIM);
  v4f x0 = __builtin_nontemporal_load(xr + lane);
  v4f x1 = __builtin_nontemporal_load(xr + lane + 32);
  v4f k0 = kr[lane];
  v4f k1 = kr[lane + 32];
  float d = x0.x*k0.x + x0.y*k0.y + x0.z*k0.z + x0.w*k0.w
          + x1.x*k1.x + x1.y*k1.y + x1.z*k1.z + x1.w*k1.w;
  #pragma unroll
  for (int off = 16; off > 0; off >>= 1) d += __shfl_xor(d, off, 32);
  if (lane == 0) {
    dot[row] = d;
    atomicMax(segmax + s, f2ord(d));
  }
}

// a = exp((dot - segmax)/sqrt(256)); segment atomic sum (wave-reduced when uniform).
__global__ void exp_sum_kernel(const int* __restrict__ seg,
                               const unsigned* __restrict__ segmax,
                               float* __restrict__ dot,
                               float* __restrict__ segsum, int N) {
  const int row = blockIdx.x * blockDim.x + threadIdx.x;
  const bool active = row < N;
  int s = active ? seg[row] : 0;
  float a = 0.0f;
  if (active) {
    const float m = ord2f(segmax[s]);
    a = __expf((dot[row] - m) * 0.0625f);   // 1/sqrt(256)
    dot[row] = a;                            // reuse buffer for 'a'
  }
  const int s0 = __shfl(s, 0, 32);
  if (__all(s == s0)) {
    #pragma unroll
    for (int off = 16; off > 0; off >>= 1) a += __shfl_xor(a, off, 32);
    if ((threadIdx.x & 31) == 0) atomicAdd(segsum + s0, a);
  } else {
    atomicAdd(segsum + s, a);
  }
}

// out[n,:] = (a[n]/(segsum[seg]+eps)) * v_mol[seg,:]
__global__ void output_kernel(const int* __restrict__ seg,
                              const float* __restrict__ a,
                              const float* __restrict__ segsum,
                              const float* __restrict__ vmol,
                              float* __restrict__ out, int N) {
  const int row = blockIdx.x * 8 + (threadIdx.x >> 5);
  if (row >= N) return;
  const int lane = threadIdx.x & 31;
  const int s = seg[row];
  const float coeff = a[row] * fast_rcp(segsum[s] + 1e-8f);
  const v4f* vr = (const v4f*)(vmol + (size_t)s * FDIM);
  v4f* o = (v4f*)(out + (size_t)row * FDIM);
  v4f v0 = vr[lane] * coeff;
  v4f v1 = vr[lane + 32] * coeff;
  __builtin_nontemporal_store(v0, o + lane);
  __builtin_nontemporal_store(v1, o + lane + 32);
}

extern "C" void kernel_launch(void* const* d_in, const int* in_sizes, int n_in,
                              void* d_out, int out_size, void* d_ws, size_t ws_size,
                              hipStream_t stream) {
  const float* x   = (const float*)d_in[0];
  const float* E   = (const float*)d_in[1];
  const int*   seg = (const int*)d_in[3];
  const float* Wq  = (const float*)d_in[4];
  const float* Wkf = (const float*)d_in[5];
  const float* bkf = (const float*)d_in[6];
  const float* Wvf = (const float*)d_in[7];
  const float* kW1 = (const float*)d_in[8];
  const float* kb1 = (const float*)d_in[9];
  const float* kW2 = (const float*)d_in[10];
  const float* kb2 = (const float*)d_in[11];
  const float* kWo = (const float*)d_in[12];
  const float* kbo = (const float*)d_in[13];
  const float* vW1 = (const float*)d_in[14];
  const float* vW2 = (const float*)d_in[15];
  const float* vWo = (const float*)d_in[16];
  float* out = (float*)d_out;

  const int N = in_sizes[0] / FDIM;   // 400000
  const int B = in_sizes[1];          // 512
  const size_t mat = (size_t)B * FDIM;

  float* ws   = (float*)d_ws;
  float* Kin  = ws;
  float* Vin  = Kin  + mat;
  float* H    = Vin  + mat;
  float* Kr   = H    + mat;
  float* Vr   = Kr   + mat;
  float* Kmol = Vr   + mat;
  float* Vmol = Kmol + mat;
  float* kp   = Vmol + mat;
  float* dot  = kp   + mat;
  unsigned* segmax = (unsigned*)(dot + N);
  float*    segsum = (float*)(segmax + B);

  init_seg_kernel<<<(B + 255) / 256, 256, 0, stream>>>(segmax, segsum, B);
  mol_embed_kernel<<<(int)((mat + 255) / 256), 256, 0, stream>>>(E, Wkf, bkf, Wvf,
                                                                 Kin, Vin, (int)mat);

  const int gemm_blocks = ((B / 16) * (FDIM / 16) + 7) / 8;
  // k path:  H = act(act(Kin)@kW1.T + kb1); Kr = Kin + (H@kW2.T + kb2); Kmol = act(Kr)@kWo.T + kbo
  gemm256_wmma_kernel<true,  true,  false, true,  false><<<gemm_blocks, 256, 0, stream>>>(Kin, kW1, kb1, nullptr, H,    B);
  gemm256_wmma_kernel<false, false, false, true,  true ><<<gemm_blocks, 256, 0, stream>>>(H,   kW2, kb2, Kin,     Kr,   B);
  gemm256_wmma_kernel<true,  false, false, true,  false><<<gemm_blocks, 256, 0, stream>>>(Kr,  kWo, kbo, nullptr, Kmol, B);
  // v path (zero biases)
  gemm256_wmma_kernel<true,  true,  false, false, false><<<gemm_blocks, 256, 0, stream>>>(Vin, vW1, nullptr, nullptr, H,    B);
  gemm256_wmma_kernel<false, false, false, false, true ><<<gemm_blocks, 256, 0, stream>>>(H,   vW2, nullptr, Vin,     Vr,   B);
  gemm256_wmma_kernel<true,  false, false, false, false><<<gemm_blocks, 256, 0, stream>>>(Vr,  vWo, nullptr, nullptr, Vmol, B);
  // kp = Kmol @ Wq   (collapses the 400000x256x256 GEMM out of the hot path)
  gemm256_wmma_kernel<false, false, true,  false, false><<<gemm_blocks, 256, 0, stream>>>(Kmol, Wq, nullptr, nullptr, kp, B);

  dot_max_kernel<<<(N + 7) / 8, 256, 0, stream>>>(x, kp, seg, dot, segmax, N);
  exp_sum_kernel<<<(N + 255) / 256, 256, 0, stream>>>(seg, segmax, dot, segsum, N);
  output_kernel<<<(N + 7) / 8, 256, 0, stream>>>(seg, dot, segsum, Vmol, out, N);
}